// GraphConvLayer_27204322853087
// MI455X (gfx1250) — compile-verified
//
#include <hip/hip_runtime.h>
#include <hip/hip_bf16.h>

// Problem constants (match reference)
#define BB 8
#define NN 1024
#define CC 256
#define C_OUT 256
#define EE 16384
#define BN (BB * NN)            // 8192
#define WORDS_PER_ROW (NN / 32) // 32 u32 words = 1024-bit adjacency row

typedef float v2f __attribute__((ext_vector_type(2)));
typedef float v4f __attribute__((ext_vector_type(4)));
typedef float v8f __attribute__((ext_vector_type(8)));
typedef int   v4i __attribute__((ext_vector_type(4)));

// Async global->LDS staging (CDNA5 GLOBAL_LOAD_ASYNC_TO_LDS_B128, ASYNCcnt).
#if defined(__gfx1250__) && __has_builtin(__builtin_amdgcn_global_load_async_to_lds_b128) && \
    __has_builtin(__builtin_amdgcn_s_wait_asynccnt)
#define USE_ASYNC_LDS 1
// Param 0 probe-confirmed: AS1 (HIP '__device__') pointer to int4, non-const.
typedef __attribute__((address_space(1))) v4i gv4i; // global int4
typedef __attribute__((address_space(3))) v4i lv4i; // LDS int4
#else
#define USE_ASYNC_LDS 0
#endif

// ---------------------------------------------------------------------------
// K1: zero the adjacency bitmap (must happen every call; scatter only ORs)
// ---------------------------------------------------------------------------
__global__ __launch_bounds__(256) void gcn_zero_bitmap(unsigned* __restrict__ bm, int n) {
    int i = blockIdx.x * blockDim.x + threadIdx.x;
    if (i < n) bm[i] = 0u;
}

// ---------------------------------------------------------------------------
// K2: scatter edges into per-node bitmap rows. atomicOr == .set() dedup.
// edge_index layout: [2, E, B] int32.  bit(src -> dst) within each graph.
// ---------------------------------------------------------------------------
__global__ __launch_bounds__(256) void gcn_scatter(const int* __restrict__ ei,
                                                   unsigned* __restrict__ bm) {
    int idx = blockIdx.x * blockDim.x + threadIdx.x; // over E*B
    if (idx >= EE * BB) return;
    int e = idx / BB;
    int b = idx - e * BB;
    int src = ei[e * BB + b];                 // edge_index[0][e][b]
    int dst = ei[EE * BB + e * BB + b];       // edge_index[1][e][b]
    unsigned* row = bm + (size_t)(b * NN + src) * WORDS_PER_ROW;
    atomicOr(&row[dst >> 5], 1u << (dst & 31));
}

// ---------------------------------------------------------------------------
// K3: degree = popcount(row) + 1 (self loop from +eye); dinv = rsqrt(deg)
// ---------------------------------------------------------------------------
__global__ __launch_bounds__(256) void gcn_degree(const unsigned* __restrict__ bm,
                                                  float* __restrict__ dinv) {
    int node = blockIdx.x * blockDim.x + threadIdx.x;
    if (node >= BN) return;
    const unsigned* row = bm + (size_t)node * WORDS_PER_ROW;
    int cnt = 1; // +eye self loop
#pragma unroll
    for (int w = 0; w < WORDS_PER_ROW; ++w) cnt += __popc(row[w]);
    dinv[node] = rsqrtf((float)cnt); // deg >= 1 always, so D>0 guard is moot
}

// ---------------------------------------------------------------------------
// K4: sparse aggregation  h[i,:] = d[i] * ( sum_{j in adj(i)} d[j]*x[j,:]
//                                           + d[i]*x[i,:] )        (+eye term)
// 4 nodes per 256-thread block; 64 threads x float4 per node row -> b128
// loads, fully coalesced 1 KB neighbor rows. Bit loop is wave-uniform.
// ---------------------------------------------------------------------------
__global__ __launch_bounds__(256) void gcn_aggregate(const float* __restrict__ x,
                                                     const unsigned* __restrict__ bm,
                                                     const float* __restrict__ dinv,
                                                     float* __restrict__ h) {
    const int node = blockIdx.x * 4 + (threadIdx.x >> 6); // global node id
    const int c4   = threadIdx.x & 63;                    // float4 slot (64*4=256)
    const int b    = node >> 10;                          // graph id (N=1024)
    const float di = dinv[node];
    const float* xb   = x + (size_t)b * NN * CC;
    const float* dv_b = dinv + b * NN;

    v4f acc = di * ((const v4f*)(x + (size_t)node * CC))[c4]; // +eye self-loop
    const unsigned* row = bm + (size_t)node * WORDS_PER_ROW;
#pragma unroll 4
    for (int w = 0; w < WORDS_PER_ROW; ++w) {
        unsigned bits = row[w]; // uniform across each wave -> no divergence cost
        while (bits) {
            int j = __ffs(bits) - 1;
            bits &= bits - 1;
            int nbr = (w << 5) + j;
            acc += dv_b[nbr] * ((const v4f*)(xb + (size_t)nbr * CC))[c4];
        }
    }
    ((v4f*)(h + (size_t)node * CC))[c4] = di * acc;
}

// ---------------------------------------------------------------------------
// K5: out = h @ W^T + bias via V_WMMA_F32_16X16X4_F32 (exact f32 path).
// Workgroup stages its 64x256 W strip (64 KB) into LDS once (async global->LDS
// B128 when available), killing the 8x per-wave redundancy; B fragments then
// come from LDS (ds_load_b64). Wave tile: 16(M) x 64(N), 4 accumulators.
// A frag (16x4):  lane L<16: h[m0+L, k0+{0,1}],  L>=16: h[m0+L-16, k0+{2,3}]
// B frag (4x16):  lane L<16: W^T[k0+{0,1}, n],   L>=16: W^T[k0+{2,3}, n]
// D/C (16x16 f32): VGPR r -> rows m0+r (lanes 0-15) and m0+r+8 (lanes 16-31).
// ---------------------------------------------------------------------------
__global__ __launch_bounds__(256) void gcn_linear_wmma(const float* __restrict__ h,
                                                       const float* __restrict__ W,
                                                       const float* __restrict__ bias,
                                                       float* __restrict__ out) {
    __shared__ float sW[64 * CC]; // 64 KB: W rows [n0, n0+64)

    const int wave = threadIdx.x >> 5;             // 0..7
    const int lane = threadIdx.x & 31;
    const int m0   = (blockIdx.x * 8 + wave) * 16; // M tile (grid.x = 64)
    const int n0   = blockIdx.y * 64;              // N strip (grid.y = 4)

    // ---- stage W[n0:n0+64, :] into LDS (contiguous 64 KB region) ----
#if USE_ASYNC_LDS
    {
        const char* gsrc = (const char*)(W + (size_t)n0 * CC);
        char*       ldst = (char*)sW;
#pragma unroll
        for (int i = 0; i < 16; ++i) { // 256 thr * 16 iters * 16 B = 64 KB
            int off = (threadIdx.x + i * 256) * 16;
            __builtin_amdgcn_global_load_async_to_lds_b128(
                (gv4i*)(gsrc + off), (lv4i*)(ldst + off), 0, 0);
        }
        __builtin_amdgcn_s_wait_asynccnt(0);
    }
#else
    {
        const v4f* g4 = (const v4f*)(W + (size_t)n0 * CC);
        v4f*       s4 = (v4f*)sW;
        for (int i = threadIdx.x; i < 64 * CC / 4; i += 256) s4[i] = g4[i];
    }
#endif
    __syncthreads();

    const int lm = lane & 15;          // row-in-tile (A) / col-in-tile (B,D)
    const int kh = (lane >> 4) << 1;   // 0 or 2: K sub-offset for this half-wave

    v8f acc[4];
#pragma unroll
    for (int t = 0; t < 4; ++t) {
        float bv = bias[n0 + t * 16 + lm]; // D element col depends only on lane
#pragma unroll
        for (int r = 0; r < 8; ++r) acc[t][r] = bv;
    }

    const float* arow = h + (size_t)(m0 + lm) * CC;
    for (int k0 = 0; k0 < CC; k0 += 4) {
        v2f a = *(const v2f*)(arow + k0 + kh);
#pragma unroll
        for (int t = 0; t < 4; ++t) {
            v2f bmv = *(const v2f*)(sW + (t * 16 + lm) * CC + k0 + kh); // ds_load_b64
#if __has_builtin(__builtin_amdgcn_wmma_f32_16x16x4_f32)
            acc[t] = __builtin_amdgcn_wmma_f32_16x16x4_f32(
                false, a, false, bmv, (short)0, acc[t], false, false);
#else
#pragma unroll
            for (int r = 0; r < 8; ++r) acc[t][r] += a.x * bmv.x + a.y * bmv.y;
#endif
        }
    }

    const int mbase = m0 + ((lane >> 4) << 3); // lanes>=16 hold rows +8
#pragma unroll
    for (int t = 0; t < 4; ++t)
#pragma unroll
        for (int r = 0; r < 8; ++r)
            out[(size_t)(mbase + r) * C_OUT + n0 + t * 16 + lm] = acc[t][r];
}

// ---------------------------------------------------------------------------
// Launch: ws layout  [0, 1MB) bitmap | [1MB, +32KB) dinv | then h (8MB)
// ---------------------------------------------------------------------------
extern "C" void kernel_launch(void* const* d_in, const int* in_sizes, int n_in,
                              void* d_out, int out_size, void* d_ws, size_t ws_size,
                              hipStream_t stream) {
    const float* x    = (const float*)d_in[0]; // [8,1024,256]
    const int*   ei   = (const int*)d_in[1];   // [2,16384,8]
    const float* W    = (const float*)d_in[2]; // [256,256]
    const float* bias = (const float*)d_in[3]; // [256]
    float*       out  = (float*)d_out;         // [8,1024,256]

    unsigned char* ws = (unsigned char*)d_ws;
    unsigned* bitmap = (unsigned*)ws;                          // 8192*32 u32 = 1 MB
    float*    dinv   = (float*)(ws + (size_t)BN * WORDS_PER_ROW * 4);
    float*    hbuf   = (float*)(ws + (size_t)BN * WORDS_PER_ROW * 4 + BN * 4);

    const int bm_words = BN * WORDS_PER_ROW; // 262144
    gcn_zero_bitmap<<<(bm_words + 255) / 256, 256, 0, stream>>>(bitmap, bm_words);
    gcn_scatter<<<(EE * BB + 255) / 256, 256, 0, stream>>>(ei, bitmap);
    gcn_degree<<<(BN + 255) / 256, 256, 0, stream>>>(bitmap, dinv);
    gcn_aggregate<<<BN / 4, 256, 0, stream>>>(x, bitmap, dinv, hbuf);

    dim3 grid(BN / 16 / 8, C_OUT / 64); // (64, 4)
    gcn_linear_wmma<<<grid, 256, 0, stream>>>(hbuf, W, bias, out);
}